// VectorQuantiser_20684562497705
// MI455X (gfx1250) — compile-verified
//
#include <hip/hip_runtime.h>

typedef __attribute__((ext_vector_type(2))) float v2f;
typedef __attribute__((ext_vector_type(8))) float v8f;

#define BATCH   16
#define CDIM    64
#define HW      4096
#define NROWS   (BATCH * HW)       // 65536
#define KEMB    1024
#define TPB     256
#define NWAVES  8
#define ROWS_WG 128                 // 8 waves * 16 rows
#define NWG     (NROWS / ROWS_WG)   // 512

// Dynamic LDS layout (floats):
//   [0, 65536)        : E in WMMA B-fragment layout
//   [65536, 66560)    : e_sq[K]
//   [66560, 66688)    : argmin indices for this WG's 128 rows (as int)
//   [66688, 66696)    : per-wave loss partials
#define LDS_FLOATS 66696
#define LDS_BYTES  (LDS_FLOATS * 4)

__global__ void vq_zero_kernel(float* acc) {
    if (threadIdx.x == 0) acc[0] = 0.0f;
}

__global__ __launch_bounds__(TPB)
void vq_main_kernel(const float* __restrict__ x,
                    const float* __restrict__ E,
                    float* __restrict__ zq,
                    float* __restrict__ lossAcc)
{
    extern __shared__ float lds[];
    float* Bfrag  = lds;            // 65536 floats
    float* esq    = lds + 65536;    // 1024 floats
    int*   idxs   = (int*)(lds + 66560); // 128 ints
    float* redbuf = lds + 66688;    // 8 floats

    const int tid = threadIdx.x;
    const int wg  = blockIdx.x;

    // ---- Stage 1: load codebook into LDS in B-fragment layout ----
    // Pair p: k = p>>5, j = p&31, c0 = 2j. Fragment target:
    //   t = k>>4 (column tile), n = k&15, kc = c0>>2, g = (c0>>1)&1
    //   float offset = ((t*16 + kc)*32 + (g*16 + n)) * 2
    for (int p = tid; p < (KEMB * CDIM / 2); p += TPB) {
        int k  = p >> 5;
        int j  = p & 31;
        int c0 = j * 2;
        float e0 = E[k * CDIM + c0];
        float e1 = E[k * CDIM + c0 + 1];
        int t  = k >> 4;
        int n  = k & 15;
        int kc = c0 >> 2;
        int g  = (c0 >> 1) & 1;
        int fo = ((t * 16 + kc) * 32 + (g * 16 + n)) * 2;
        Bfrag[fo]     = e0;
        Bfrag[fo + 1] = e1;
    }
    // e_sq[k] = sum_c E[k][c]^2
    for (int k = tid; k < KEMB; k += TPB) {
        float s = 0.0f;
        for (int c = 0; c < CDIM; ++c) {
            float e = E[k * CDIM + c];
            s += e * e;
        }
        esq[k] = s;
    }
    __syncthreads();

    // ---- Stage 2: WMMA distances + fused argmin ----
    const int wave = tid >> 5;
    const int lane = tid & 31;
    const int m    = lane & 15;     // row within 16-row tile
    const int g    = lane >> 4;     // half-wave selects K pair {0,1} vs {2,3}
    const int rowbase = wg * ROWS_WG + wave * 16;
    const int b       = rowbase >> 12;        // image index (128 | 4096)
    const int nrow    = (rowbase & 4095) + m; // spatial index h*w

    // A fragments: 16 chunks of K=4; per ISA 16x4 f32 A layout lane holds
    // (c0, c0+1) with c0 = kc*4 + 2*g.  x_flat[row][c] = x[(b*C + c)*HW + n]
    v2f afr[16];
#pragma unroll
    for (int kc = 0; kc < 16; ++kc) {
        int c0 = kc * 4 + g * 2;
        afr[kc].x = x[(b * CDIM + c0    ) * HW + nrow];
        afr[kc].y = x[(b * CDIM + c0 + 1) * HW + nrow];
    }

    float minval[8];
    int   minidx[8];
#pragma unroll
    for (int r = 0; r < 8; ++r) { minval[r] = 3.402823466e38f; minidx[r] = 0; }

    for (int t = 0; t < KEMB / 16; ++t) {
        v8f acc = {0.f, 0.f, 0.f, 0.f, 0.f, 0.f, 0.f, 0.f};
        const v2f* bp = (const v2f*)(Bfrag + t * 1024);
#pragma unroll
        for (int kc = 0; kc < 16; ++kc) {
            v2f bf = bp[kc * 32 + lane];   // conflict-free ds_load_b64
            acc = __builtin_amdgcn_wmma_f32_16x16x4_f32(
                false, afr[kc], false, bf, (short)0, acc, false, false);
        }
        int   col = t * 16 + m;
        float es  = esq[col];
#pragma unroll
        for (int r = 0; r < 8; ++r) {
            float d = es - 2.0f * acc[r];   // ||x||^2 constant per row: drop it
            if (d < minval[r]) { minval[r] = d; minidx[r] = col; }
        }
    }

    // Reduce across the 16 lanes that share a row (xor within half-wave).
#pragma unroll
    for (int r = 0; r < 8; ++r) {
        float v = minval[r];
        int   i = minidx[r];
#pragma unroll
        for (int off = 1; off <= 8; off <<= 1) {
            float ov = __shfl_xor(v, off, 32);
            int   oi = __shfl_xor(i, off, 32);
            if (ov < v || (ov == v && oi < i)) { v = ov; i = oi; }
        }
        // D layout: VGPR r holds row r (lanes 0-15) and row r+8 (lanes 16-31)
        if (m == 0) idxs[wave * 16 + g * 8 + r] = i;
    }
    __syncthreads();

    // ---- Stage 3: gather z_q from LDS codebook + fused loss ----
    const int b_wg = (wg * ROWS_WG) >> 12;
    const int n_wg = (wg * ROWS_WG) & 4095;
    float lsum = 0.0f;
    for (int ii = tid; ii < ROWS_WG * CDIM; ii += TPB) {
        int rl = ii & 127;          // local row -> coalesced along n
        int c  = ii >> 7;
        int k  = idxs[rl];
        int t  = k >> 4, nn = k & 15, kc = c >> 2, gg = (c >> 1) & 1;
        float ev = Bfrag[((t * 16 + kc) * 32 + (gg * 16 + nn)) * 2 + (c & 1)];
        int gaddr = (b_wg * CDIM + c) * HW + n_wg + rl;
        float xv = x[gaddr];
        zq[gaddr] = ev;
        float dlt = ev - xv;
        lsum += dlt * dlt;
    }
#pragma unroll
    for (int off = 1; off < 32; off <<= 1) lsum += __shfl_xor(lsum, off, 32);
    if (lane == 0) redbuf[wave] = lsum;
    __syncthreads();
    if (tid == 0) {
        float tot = 0.0f;
        for (int wv = 0; wv < NWAVES; ++wv) tot += redbuf[wv];
        atomicAdd(lossAcc, tot);
    }
}

__global__ void vq_finalize_kernel(const float* acc, float* lossOut) {
    if (threadIdx.x == 0) {
        // loss = mean((sg(zq)-x)^2) + mean((zq-sg(x))^2) = 2 * mean((zq-x)^2)
        lossOut[0] = 2.0f * acc[0] / (float)((long)NROWS * CDIM);
    }
}

extern "C" void kernel_launch(void* const* d_in, const int* in_sizes, int n_in,
                              void* d_out, int out_size, void* d_ws, size_t ws_size,
                              hipStream_t stream) {
    const float* x = (const float*)d_in[0];      // [16,64,64,64] f32
    const float* E = (const float*)d_in[1];      // [1024,64] f32
    float* out     = (float*)d_out;              // z_q (4194304) + loss (1)
    float* acc     = (float*)d_ws;               // loss accumulator

    vq_zero_kernel<<<1, 64, 0, stream>>>(acc);
    vq_main_kernel<<<NWG, TPB, LDS_BYTES, stream>>>(x, E, out, acc);
    vq_finalize_kernel<<<1, 64, 0, stream>>>(acc, out + (out_size - 1));
}